// NavierStokes_13778255086286
// MI455X (gfx1250) — compile-verified
//
#include <hip/hip_runtime.h>
#include <hip/hip_bf16.h>

typedef float v2f __attribute__((ext_vector_type(2)));
typedef float v8f __attribute__((ext_vector_type(8)));

#define NCH   13
#define RS    18              // LDS row stride (floats): 8*18 % 64 == 16 -> halves hit disjoint banks
#define CROWS 28              // rows 0..19 real, 24..27 slack written by tile1 upper half
#define CST   (CROWS * RS)    // floats per channel slab

__device__ __forceinline__ v8f wmma4(v2f a, v2f b, v8f c) {
  // D = A(16x4,f32) * B(4x16,f32) + C(16x16,f32), full fp32 matrix pipe
  return __builtin_amdgcn_wmma_f32_16x16x4_f32(false, a, false, b, (short)0, c, false, false);
}

struct F12 { float v[12]; };

__device__ __forceinline__ int rowOf(int j, int half) {
  // j 0..7 -> D tile0 vgpr j (rows j / j+8); j 8..11 -> tile1 vgpr j-8 (rows 16+(j-8) / 24+(j-8))
  return ((j < 8) ? j : j + 8) + 8 * half;
}

__device__ __forceinline__ void st12(float* chbase, int p, int half, const F12& x) {
#pragma unroll
  for (int j = 0; j < 12; ++j)
    chbase[rowOf(j, half) * RS + p] = x.v[j];
}

__device__ __forceinline__ F12 extract12(v8f c0, v8f c1) {
  F12 r;
#pragma unroll
  for (int j = 0; j < 8; ++j) r.v[j] = c0[j];
#pragma unroll
  for (int j = 0; j < 4; ++j) r.v[8 + j] = c1[j];
  return r;
}

__device__ __forceinline__ float fast_tanh(float s) {
  // tanh(|s|) = (1 - e) / (1 + e), e = exp(-2|s|); v_exp_f32 + v_rcp_f32, no IEEE div expansion
  float e = __expf(-2.f * fabsf(s));
  float a = (1.f - e) * __builtin_amdgcn_rcpf(1.f + e);
  return copysignf(a, s);
}

// Propagate the 13-channel 3rd-order jet through one tanh layer.
// getS(c, out): pre-activation S_c = W @ h_c in "F12" (D-layout useful rows).
template <class GetS>
__device__ __forceinline__ void layer_core(float* sm, int p, int half,
                                           const float* bias, GetS getS) {
  float f1[12], f2[12], f3[12];
  {
    F12 S0; getS(0, S0);
    F12 a;
#pragma unroll
    for (int j = 0; j < 12; ++j) {
      float s  = S0.v[j] + bias[j];
      float av = fast_tanh(s);
      float t1 = 1.f - av * av;           // phi'
      f1[j] = t1;
      f2[j] = -2.f * av * t1;             // phi''
      f3[j] = t1 * (6.f * av * av - 2.f); // phi'''
      a.v[j] = av;
    }
    st12(sm, p, half, a);
  }
  F12 Sx; getS(1, Sx);
  { F12 o;
#pragma unroll
    for (int j = 0; j < 12; ++j) o.v[j] = f1[j] * Sx.v[j];
    st12(sm + 1 * CST, p, half, o); }
  F12 Sy; getS(2, Sy);
  { F12 o;
#pragma unroll
    for (int j = 0; j < 12; ++j) o.v[j] = f1[j] * Sy.v[j];
    st12(sm + 2 * CST, p, half, o); }
  F12 St; getS(3, St);
  { F12 o;
#pragma unroll
    for (int j = 0; j < 12; ++j) o.v[j] = f1[j] * St.v[j];
    st12(sm + 3 * CST, p, half, o); }
  F12 Sxx; getS(4, Sxx);
  { F12 o;
#pragma unroll
    for (int j = 0; j < 12; ++j)
      o.v[j] = f2[j] * Sx.v[j] * Sx.v[j] + f1[j] * Sxx.v[j];
    st12(sm + 4 * CST, p, half, o); }
  F12 Sxy; getS(5, Sxy);
  { F12 o;
#pragma unroll
    for (int j = 0; j < 12; ++j)
      o.v[j] = f2[j] * Sx.v[j] * Sy.v[j] + f1[j] * Sxy.v[j];
    st12(sm + 5 * CST, p, half, o); }
  F12 Syy; getS(6, Syy);
  { F12 o;
#pragma unroll
    for (int j = 0; j < 12; ++j)
      o.v[j] = f2[j] * Sy.v[j] * Sy.v[j] + f1[j] * Syy.v[j];
    st12(sm + 6 * CST, p, half, o); }
  { F12 T; getS(7, T); F12 o;
#pragma unroll
    for (int j = 0; j < 12; ++j)
      o.v[j] = f2[j] * Sx.v[j] * St.v[j] + f1[j] * T.v[j];
    st12(sm + 7 * CST, p, half, o); }
  { F12 T; getS(8, T); F12 o;
#pragma unroll
    for (int j = 0; j < 12; ++j)
      o.v[j] = f2[j] * Sy.v[j] * St.v[j] + f1[j] * T.v[j];
    st12(sm + 8 * CST, p, half, o); }
  { F12 T; getS(9, T); F12 o;
#pragma unroll
    for (int j = 0; j < 12; ++j)
      o.v[j] = f3[j] * Sx.v[j] * Sx.v[j] * Sx.v[j]
             + 3.f * f2[j] * Sxx.v[j] * Sx.v[j] + f1[j] * T.v[j];
    st12(sm + 9 * CST, p, half, o); }
  { F12 T; getS(10, T); F12 o;
#pragma unroll
    for (int j = 0; j < 12; ++j)
      o.v[j] = f3[j] * Sx.v[j] * Sx.v[j] * Sy.v[j]
             + f2[j] * (Sxx.v[j] * Sy.v[j] + 2.f * Sxy.v[j] * Sx.v[j])
             + f1[j] * T.v[j];
    st12(sm + 10 * CST, p, half, o); }
  { F12 T; getS(11, T); F12 o;
#pragma unroll
    for (int j = 0; j < 12; ++j)
      o.v[j] = f3[j] * Sx.v[j] * Sy.v[j] * Sy.v[j]
             + f2[j] * (Syy.v[j] * Sx.v[j] + 2.f * Sxy.v[j] * Sy.v[j])
             + f1[j] * T.v[j];
    st12(sm + 11 * CST, p, half, o); }
  { F12 T; getS(12, T); F12 o;
#pragma unroll
    for (int j = 0; j < 12; ++j)
      o.v[j] = f3[j] * Sy.v[j] * Sy.v[j] * Sy.v[j]
             + 3.f * f2[j] * Syy.v[j] * Sy.v[j] + f1[j] * T.v[j];
    st12(sm + 12 * CST, p, half, o); }
}

__global__ __launch_bounds__(32) void pinn_ns_kernel(
    const float* __restrict__ gx, const float* __restrict__ gy, const float* __restrict__ gt,
    const float* __restrict__ W0, const float* __restrict__ b0,
    const float* __restrict__ Wh, const float* __restrict__ bh,
    const float* __restrict__ W9, const float* __restrict__ b9,
    const float* __restrict__ lam1, const float* __restrict__ lam2,
    float* __restrict__ out, int n)
{
  __shared__ float sm[NCH * CST];
  const int lane = threadIdx.x;        // wave32: one wave per block
  const int p    = lane & 15;          // point within group (WMMA N / lane)
  const int half = lane >> 4;
  const int base = blockIdx.x * 16;
  if (base >= n) return;
  const int idx  = (base + p < n) ? (base + p) : (n - 1);

  float xv = gx[idx], yv = gy[idx], tv = gt[idx];

  // ---------------- layer 0: W0 (20x3), K padded to 4, 1 K-step, 2 M tiles ----
  v2f A0[2];
  {
    int c0 = 2 * half, c1 = 1 + 2 * half;
    int r1 = 16 + p;
    A0[0].x = (c0 < 3) ? W0[p * 3 + c0] : 0.f;
    A0[0].y = (c1 < 3) ? W0[p * 3 + c1] : 0.f;
    A0[1].x = (r1 < 20 && c0 < 3) ? W0[r1 * 3 + c0] : 0.f;
    A0[1].y = (r1 < 20 && c1 < 3) ? W0[r1 * 3 + c1] : 0.f;
  }
  float bias[12];
#pragma unroll
  for (int j = 0; j < 12; ++j) {
    int row = rowOf(j, half);
    bias[j] = (row < 20) ? b0[row] : 0.f;
  }
  {
    auto getS = [&](int c, F12& r) {
      v2f B;
      if (c == 0)      { B.x = half ? tv  : xv;  B.y = half ? 0.f : yv; }
      else if (c == 1) { B.x = half ? 0.f : 1.f; B.y = 0.f; }
      else if (c == 2) { B.x = 0.f;              B.y = half ? 0.f : 1.f; }
      else if (c == 3) { B.x = half ? 1.f : 0.f; B.y = 0.f; }
      else {
#pragma unroll
        for (int j = 0; j < 12; ++j) r.v[j] = 0.f;
        return;
      }
      v8f c0 = {}; v8f c1 = {};
      c0 = wmma4(A0[0], B, c0);
      c1 = wmma4(A0[1], B, c1);
      r = extract12(c0, c1);
    };
    layer_core(sm, p, half, bias, getS);
  }
  __syncthreads();

  // ---------------- 8 hidden layers: W (20x20), 5 K-steps, 2 M tiles ----------
#pragma unroll 1
  for (int i = 0; i < 8; ++i) {
    const float* W = Wh + i * 400;
    const float* b = bh + i * 20;
    v2f AW[2][5];
#pragma unroll
    for (int tt = 0; tt < 2; ++tt)
#pragma unroll
      for (int k = 0; k < 5; ++k) {
        int row = tt * 16 + p;
        int cc  = 4 * k + 2 * half;
        AW[tt][k].x = (row < 20) ? W[row * 20 + cc] : 0.f;
        AW[tt][k].y = (row < 20) ? W[row * 20 + cc + 1] : 0.f;
      }
#pragma unroll
    for (int j = 0; j < 12; ++j) {
      int row = rowOf(j, half);
      bias[j] = (row < 20) ? b[row] : 0.f;
    }
    auto getS = [&](int c, F12& r) {
      const float* hb = sm + c * CST;
      v2f B[5];
#pragma unroll
      for (int k = 0; k < 5; ++k) {
        B[k].x = hb[(4 * k + 2 * half) * RS + p];
        B[k].y = hb[(4 * k + 1 + 2 * half) * RS + p];
      }
      v8f c0 = {}; v8f c1 = {};
#pragma unroll
      for (int k = 0; k < 5; ++k) {
        c0 = wmma4(AW[0][k], B[k], c0);
        c1 = wmma4(AW[1][k], B[k], c1);
      }
      r = extract12(c0, c1);
    };
    layer_core(sm, p, half, bias, getS);
    __syncthreads();
  }

  // ---------------- output layer: W9 (2x20), 1 M tile, 5 K-steps --------------
  v2f A9[5];
#pragma unroll
  for (int k = 0; k < 5; ++k) {
    int cc = 4 * k + 2 * half;
    A9[k].x = (p < 2) ? W9[p * 20 + cc] : 0.f;
    A9[k].y = (p < 2) ? W9[p * 20 + cc + 1] : 0.f;
  }
  float o0[13], o1[13];   // row0 = psi-jet, row1 = p-jet (valid in lanes 0..15)
#pragma unroll
  for (int c = 0; c < 13; ++c) {
    const float* hb = sm + c * CST;
    v8f acc = {};
#pragma unroll
    for (int k = 0; k < 5; ++k) {
      v2f B;
      B.x = hb[(4 * k + 2 * half) * RS + p];
      B.y = hb[(4 * k + 1 + 2 * half) * RS + p];
      acc = wmma4(A9[k], B, acc);
    }
    o0[c] = acc[0];
    o1[c] = acc[1];
  }

  // ---------------- per-lane NS residual combine ------------------------------
  if (lane < 16 && base + p < n) {
    float l1 = lam1[0], l2 = lam2[0];
    float u    =  o0[2];    // psi_y
    float v_   = -o0[1];    // -psi_x
    float pp   =  o1[0] + b9[1];
    float u_x  =  o0[5];    // psi_xy
    float u_y  =  o0[6];    // psi_yy
    float u_t  =  o0[8];    // psi_yt
    float u_xx =  o0[10];   // psi_xxy
    float u_yy =  o0[12];   // psi_yyy
    float v_x  = -o0[4];    // -psi_xx
    float v_y  = -o0[5];    // -psi_xy
    float v_t  = -o0[7];    // -psi_xt
    float v_xx = -o0[9];    // -psi_xxx
    float v_yy = -o0[11];   // -psi_xyy
    float p_x  =  o1[1], p_y = o1[2];
    float f = u_t + l1 * (u * u_x + v_ * u_y) + p_x - l2 * (u_xx + u_yy);
    float g = v_t + l1 * (u * v_x + v_ * v_y) + p_y - l2 * (v_xx + v_yy);
    float* o = out + (size_t)(base + p) * 5;
    o[0] = u; o[1] = v_; o[2] = pp; o[3] = f; o[4] = g;
  }
}

extern "C" void kernel_launch(void* const* d_in, const int* in_sizes, int n_in,
                              void* d_out, int out_size, void* d_ws, size_t ws_size,
                              hipStream_t stream) {
  const float* x  = (const float*)d_in[0];
  const float* y  = (const float*)d_in[1];
  const float* t  = (const float*)d_in[2];
  const float* W0 = (const float*)d_in[3];
  const float* b0 = (const float*)d_in[4];
  const float* Wh = (const float*)d_in[5];
  const float* bh = (const float*)d_in[6];
  const float* W9 = (const float*)d_in[7];
  const float* b9 = (const float*)d_in[8];
  const float* l1 = (const float*)d_in[9];
  const float* l2 = (const float*)d_in[10];
  int n = in_sizes[0];
  int nblk = (n + 15) / 16;
  pinn_ns_kernel<<<nblk, 32, 0, stream>>>(x, y, t, W0, b0, Wh, bh, W9, b9, l1, l2,
                                          (float*)d_out, n);
}